// DE_GCL_vel_2D_10599979287282
// MI455X (gfx1250) — compile-verified
//
#include <hip/hip_runtime.h>

typedef __attribute__((ext_vector_type(2))) float v2f;
typedef __attribute__((ext_vector_type(8))) float v8f;

#define NNODES 50000
#define NEDGES 800000
#define NFD 128
#define HD 128

// D(16x16) += A(16x4) * B(4x16) accumulated over K, fp32 WMMA.
// A: LDS tile, row-major 16 x K (stride lda). B: global, K x ldb row-major, cols [n0, n0+16).
__device__ __forceinline__ v8f wmma_f32_acc(const float* __restrict__ A_lds, int lda,
                                            const float* __restrict__ B_g, int ldb,
                                            int n0, int K, v8f acc) {
  const int lane = threadIdx.x & 31;
  const int h2 = (lane >> 4) << 1;   // 0 or 2 (K offset for lane half)
  const int mn = lane & 15;
  for (int k = 0; k < K; k += 4) {
    v2f a, b;
    a.x = A_lds[mn * lda + k + h2];
    a.y = A_lds[mn * lda + k + h2 + 1];
    b.x = B_g[(k + h2) * ldb + n0 + mn];
    b.y = B_g[(k + h2 + 1) * ldb + n0 + mn];
    acc = __builtin_amdgcn_wmma_f32_16x16x4_f32(false, a, false, b, (short)0, acc, false, false);
  }
  return acc;
}

__global__ void zero_kernel(float* __restrict__ p, int n) {
  int i = blockIdx.x * blockDim.x + threadIdx.x;
  if (i < n) p[i] = 0.f;
}

// Per 16-node tile: hr = h@W_hr + be1, hc = h@W_hc, vs = relu(h@Wv1+bv1)@Wv2 + bv2
__global__ __launch_bounds__(256) void node_pre_kernel(
    const float* __restrict__ h, const float* __restrict__ We1,
    const float* __restrict__ be1,
    const float* __restrict__ Wv1, const float* __restrict__ bv1,
    const float* __restrict__ Wv2, const float* __restrict__ bv2,
    float* __restrict__ hr, float* __restrict__ hc, float* __restrict__ vs) {
  __shared__ float hs[16 * NFD];
  __shared__ float vss[16];
  const int t = threadIdx.x;
  const int base = blockIdx.x * 16;
  for (int idx = t; idx < 16 * NFD; idx += 256) {
    int m = idx >> 7, k = idx & 127;
    hs[idx] = h[(base + m) * NFD + k];
  }
  if (t < 16) vss[t] = bv2[0];
  __syncthreads();

  const int wave = t >> 5, lane = t & 31;
  const int n0 = wave * 16, mn = lane & 15, hf = lane >> 4;
  const v8f zero = {0.f, 0.f, 0.f, 0.f, 0.f, 0.f, 0.f, 0.f};

  v8f d = wmma_f32_acc(hs, NFD, We1, HD, n0, NFD, zero);          // W_hr = We1 rows 0..127
#pragma unroll
  for (int r = 0; r < 8; ++r)
    hr[(base + r + hf * 8) * HD + n0 + mn] = d[r] + be1[n0 + mn];

  d = wmma_f32_acc(hs, NFD, We1 + NFD * HD, HD, n0, NFD, zero);   // W_hc = rows 128..255
#pragma unroll
  for (int r = 0; r < 8; ++r)
    hc[(base + r + hf * 8) * HD + n0 + mn] = d[r];

  d = wmma_f32_acc(hs, NFD, Wv1, HD, n0, NFD, zero);
  float w2 = Wv2[n0 + mn];
  float b1 = bv1[n0 + mn];
#pragma unroll
  for (int r = 0; r < 8; ++r) {
    float u = d[r] + b1;
    u = u > 0.f ? u : 0.f;
    atomicAdd(&vss[r + hf * 8], u * w2);
  }
  __syncthreads();
  if (t < 16) vs[base + t] = vss[t];
}

// Per 16-edge tile: 4-group edge MLP (WMMA, register-cached We2 slice,
// double-buffered z tile), Wc1 WMMA, scatter agg/cnt/aggn.
__global__ __launch_bounds__(256) void edge_kernel(
    const int* __restrict__ eidx, const float* __restrict__ coord,
    const float* __restrict__ vel,
    const float* __restrict__ We1, const float* __restrict__ We2,
    const float* __restrict__ be2,
    const float* __restrict__ Wc1, const float* __restrict__ bc1,
    const float* __restrict__ Wc2,
    const float* __restrict__ hr, const float* __restrict__ hc,
    float* __restrict__ agg, float* __restrict__ cnt, float* __restrict__ aggn) {
  __shared__ float base_s[16 * HD];
  __shared__ float xt_s[16 * HD];
  __shared__ float yt_s[16 * HD];
  __shared__ float z_s[2][16 * HD];
  __shared__ int rowi[16], coli[16];
  __shared__ float crx[16], cry[16], ccx[16], ccy[16];
  __shared__ float vrx[16], vry[16], vcx[16], vcy[16];
  __shared__ float d2s[16];
  __shared__ float cm_s[32];

  const int t = threadIdx.x;
  const int e0 = blockIdx.x * 16;
  if (t < 16) {
    int e = e0 + t;
    int r = eidx[e];
    int c = eidx[NEDGES + e];
    rowi[t] = r; coli[t] = c;
    float ax = coord[2 * r], ay = coord[2 * r + 1];
    float bx = coord[2 * c], by = coord[2 * c + 1];
    crx[t] = ax; cry[t] = ay; ccx[t] = bx; ccy[t] = by;
    vrx[t] = vel[2 * r]; vry[t] = vel[2 * r + 1];
    vcx[t] = vel[2 * c]; vcy[t] = vel[2 * c + 1];
    float dx = ax - bx, dy = ay - by;
    d2s[t] = dx * dx + dy * dy;
  }
  if (t < 32) cm_s[t] = 0.f;
  __syncthreads();

  const float* Wr0 = We1 + 256 * HD;  // radial: cgr.x
  const float* Wr1 = We1 + 257 * HD;  // cgr.y
  const float* Wr2 = We1 + 258 * HD;  // cgc.x
  const float* Wr3 = We1 + 259 * HD;  // cgc.y
  const float* Wr4 = We1 + 260 * HD;  // |d|^2 (group-invariant)
  const float* Wq0 = We1 + 261 * HD;  // vgr.x
  const float* Wq1 = We1 + 262 * HD;  // vgr.y
  const float* Wq2 = We1 + 263 * HD;  // vgc.x
  const float* Wq3 = We1 + 264 * HD;  // vgc.y

  for (int idx = t; idx < 16 * HD; idx += 256) {
    int m = idx >> 7, k = idx & 127;
    float pre = hr[rowi[m] * HD + k] + hc[coli[m] * HD + k];  // includes be1
    base_s[idx] = pre + d2s[m] * Wr4[k];
    xt_s[idx] = crx[m] * Wr0[k] + ccx[m] * Wr2[k] + vrx[m] * Wq0[k] + vcx[m] * Wq2[k];
    yt_s[idx] = cry[m] * Wr1[k] + ccy[m] * Wr3[k] + vry[m] * Wq1[k] + vcy[m] * Wq3[k];
  }

  const int wave = t >> 5, lane = t & 31;
  const int n0 = wave * 16, mn = lane & 15, hf = lane >> 4;
  const v8f zero = {0.f, 0.f, 0.f, 0.f, 0.f, 0.f, 0.f, 0.f};
  v8f ef = zero;
  const float sxs[4] = {1.f, -1.f, -1.f, 1.f};
  const float sys[4] = {1.f, -1.f, 1.f, -1.f};
  const float b2 = be2[n0 + mn];
  const int h2 = hf << 1;

  // Register-cache this wave's 16-column slice of We2 (K=128 -> 64 VGPRs),
  // reused across all 4 group iterations.
  v2f bw[32];
#pragma unroll
  for (int kk = 0; kk < 32; ++kk) {
    bw[kk].x = We2[(4 * kk + h2) * HD + n0 + mn];
    bw[kk].y = We2[(4 * kk + h2 + 1) * HD + n0 + mn];
  }

  __syncthreads();  // base/xt/yt tiles ready

  for (int g = 0; g < 4; ++g) {
    float sx = sxs[g], sy = sys[g];
    float* zb = z_s[g & 1];
    for (int idx = t; idx < 16 * HD; idx += 256) {
      float v = base_s[idx] + sx * xt_s[idx] + sy * yt_s[idx];
      zb[idx] = v > 0.f ? v : 0.f;
    }
    __syncthreads();  // z tile ready; prior readers of this buffer ordered by
                      // the previous iteration's barrier (double buffering)
    v8f d = zero;
#pragma unroll
    for (int kk = 0; kk < 32; ++kk) {
      v2f a;
      a.x = zb[mn * HD + 4 * kk + h2];
      a.y = zb[mn * HD + 4 * kk + h2 + 1];
      d = __builtin_amdgcn_wmma_f32_16x16x4_f32(false, a, false, bw[kk], (short)0, d,
                                                false, false);
    }
#pragma unroll
    for (int r = 0; r < 8; ++r) {
      float u = d[r] + b2;
      ef[r] += (u > 0.f ? u : 0.f);
    }
  }

  // edge_feat = ef/4; stash tile in z_s[0], scatter into aggn
#pragma unroll
  for (int r = 0; r < 8; ++r) {
    int m = r + hf * 8;
    float v = ef[r] * 0.25f;
    z_s[0][m * HD + n0 + mn] = v;
    atomicAdd(&aggn[rowi[m] * HD + n0 + mn], v);
  }
  __syncthreads();

  // cm = relu(edge_feat@Wc1 + bc1) @ Wc2   (H -> 2, lane-partial into LDS)
  v8f d = wmma_f32_acc(z_s[0], HD, Wc1, HD, n0, HD, zero);
  const float bb = bc1[n0 + mn];
  const float w20 = Wc2[(n0 + mn) * 2 + 0];
  const float w21 = Wc2[(n0 + mn) * 2 + 1];
#pragma unroll
  for (int r = 0; r < 8; ++r) {
    int m = r + hf * 8;
    float u = d[r] + bb;
    u = u > 0.f ? u : 0.f;
    atomicAdd(&cm_s[m * 2 + 0], u * w20);
    atomicAdd(&cm_s[m * 2 + 1], u * w21);
  }
  __syncthreads();

  if (t < 16) {
    float dx = crx[t] - ccx[t];
    float dy = cry[t] - ccy[t];
    float tx = fminf(fmaxf(dx * cm_s[2 * t], -100.f), 100.f);
    float ty = fminf(fmaxf(dy * cm_s[2 * t + 1], -100.f), 100.f);
    int r = rowi[t];
    atomicAdd(&agg[2 * r], tx);
    atomicAdd(&agg[2 * r + 1], ty);
    atomicAdd(&cnt[r], 1.0f);
  }
}

// Per 16-node tile: h_new = h + relu([h,aggn]@Wn1+bn1)@Wn2+bn2 ; coord update
__global__ __launch_bounds__(256) void node_post_kernel(
    const float* __restrict__ h, const float* __restrict__ coord,
    const float* __restrict__ vel,
    const float* __restrict__ Wn1, const float* __restrict__ bn1,
    const float* __restrict__ Wn2, const float* __restrict__ bn2,
    const float* __restrict__ aggn, const float* __restrict__ agg,
    const float* __restrict__ cnt, const float* __restrict__ vs,
    float* __restrict__ h_out, float* __restrict__ coord_out) {
  __shared__ float a_s[16 * 256];
  __shared__ float t_s[16 * HD];
  const int t = threadIdx.x;
  const int base = blockIdx.x * 16;
  for (int idx = t; idx < 16 * 256; idx += 256) {
    int m = idx >> 8, k = idx & 255;
    a_s[idx] = (k < NFD) ? h[(base + m) * NFD + k]
                         : aggn[(base + m) * HD + (k - NFD)];
  }
  __syncthreads();

  const int wave = t >> 5, lane = t & 31;
  const int n0 = wave * 16, mn = lane & 15, hf = lane >> 4;
  const v8f zero = {0.f, 0.f, 0.f, 0.f, 0.f, 0.f, 0.f, 0.f};

  v8f d = wmma_f32_acc(a_s, 256, Wn1, HD, n0, 256, zero);
  const float b1 = bn1[n0 + mn];
#pragma unroll
  for (int r = 0; r < 8; ++r) {
    int m = r + hf * 8;
    float u = d[r] + b1;
    t_s[m * HD + n0 + mn] = u > 0.f ? u : 0.f;
  }
  __syncthreads();

  d = wmma_f32_acc(t_s, HD, Wn2, NFD, n0, HD, zero);
  const float b2 = bn2[n0 + mn];
#pragma unroll
  for (int r = 0; r < 8; ++r) {
    int m = r + hf * 8;
    int n = n0 + mn;
    h_out[(base + m) * NFD + n] = h[(base + m) * NFD + n] + d[r] + b2;
  }

  if (t < 16) {
    int node = base + t;
    float c = cnt[node];
    float cd = c < 1.f ? 1.f : c;
    coord_out[2 * node] = coord[2 * node] + agg[2 * node] / cd + vs[node] * vel[2 * node];
    coord_out[2 * node + 1] = coord[2 * node + 1] + agg[2 * node + 1] / cd + vs[node] * vel[2 * node + 1];
  }
}

extern "C" void kernel_launch(void* const* d_in, const int* in_sizes, int n_in,
                              void* d_out, int out_size, void* d_ws, size_t ws_size,
                              hipStream_t stream) {
  const float* h = (const float*)d_in[0];
  const float* coord = (const float*)d_in[1];
  const float* vel = (const float*)d_in[2];
  const float* We1 = (const float*)d_in[3];
  const float* be1 = (const float*)d_in[4];
  const float* We2 = (const float*)d_in[5];
  const float* be2 = (const float*)d_in[6];
  const float* Wc1 = (const float*)d_in[7];
  const float* bc1 = (const float*)d_in[8];
  const float* Wc2 = (const float*)d_in[9];
  const float* Wn1 = (const float*)d_in[10];
  const float* bn1 = (const float*)d_in[11];
  const float* Wn2 = (const float*)d_in[12];
  const float* bn2 = (const float*)d_in[13];
  const float* Wv1 = (const float*)d_in[14];
  const float* bv1 = (const float*)d_in[15];
  const float* Wv2 = (const float*)d_in[16];
  const float* bv2 = (const float*)d_in[17];
  const int* eidx = (const int*)d_in[18];

  float* ws = (float*)d_ws;
  float* hr = ws;                              // N*128
  float* hc = hr + (size_t)NNODES * HD;        // N*128
  float* vs = hc + (size_t)NNODES * HD;        // N
  float* agg = vs + NNODES;                    // N*2
  float* cnt = agg + 2 * NNODES;               // N
  float* aggn = cnt + NNODES;                  // N*128

  // agg, cnt, aggn are contiguous: N*(2+1+128) floats to zero
  int nz = NNODES * (HD + 3);
  zero_kernel<<<(nz + 255) / 256, 256, 0, stream>>>(agg, nz);

  node_pre_kernel<<<NNODES / 16, 256, 0, stream>>>(h, We1, be1, Wv1, bv1, Wv2, bv2,
                                                   hr, hc, vs);
  edge_kernel<<<NEDGES / 16, 256, 0, stream>>>(eidx, coord, vel, We1, We2, be2,
                                               Wc1, bc1, Wc2, hr, hc, agg, cnt, aggn);

  float* h_out = (float*)d_out;
  float* coord_out = h_out + (size_t)NNODES * NFD;
  node_post_kernel<<<NNODES / 16, 256, 0, stream>>>(h, coord, vel, Wn1, bn1, Wn2, bn2,
                                                    aggn, agg, cnt, vs, h_out, coord_out);
}